// SocialPooling_90477781057850
// MI455X (gfx1250) — compile-verified
//
#include <hip/hip_runtime.h>

typedef __attribute__((ext_vector_type(2))) float v2f;
typedef __attribute__((ext_vector_type(8))) float v8f;

#define G_    4
#define NS_   2.0f
#define H_    128
#define POOL_ 128
#define NAG_  1024
#define GG_   (G_ * G_)
#define FANIN_ (GG_ * H_)   // 2048

__device__ __forceinline__ int finite_f(float x) {
    return (__float_as_uint(x) & 0x7F800000u) != 0x7F800000u;
}

// ---------------------------------------------------------------------------
// Kernel 0: per-agent validity masks
//   fpos[j] = all(isfinite(pos[j]))
//   vj[j]   = fpos[j] && all(isfinite(hidden[j,:]))
// ---------------------------------------------------------------------------
__global__ void validity_kernel(const float* __restrict__ pos,
                                const float* __restrict__ hidden,
                                int* __restrict__ fpos,
                                int* __restrict__ vj) {
    int j = blockIdx.x * blockDim.x + threadIdx.x;
    if (j >= NAG_) return;
    int fp = finite_f(pos[2 * j]) & finite_f(pos[2 * j + 1]);
    int vh = 1;
    const float* hrow = hidden + (size_t)j * H_;
#pragma unroll 8
    for (int k = 0; k < H_; ++k) vh &= finite_f(hrow[k]);
    fpos[j] = fp;
    vj[j]   = fp & vh;
}

// ---------------------------------------------------------------------------
// Kernel 1: social max-pool into grid
//   One block per agent i, 128 threads = one per hidden dim.
//   LDS maxbuf[16][128] (-inf init) + touched[16] flags; untouched cells -> 0.
// ---------------------------------------------------------------------------
__global__ void pool_kernel(const float* __restrict__ pos,
                            const float* __restrict__ hidden,
                            const int* __restrict__ fpos,
                            const int* __restrict__ vj,
                            float* __restrict__ pooled) {
    __shared__ float maxbuf[GG_ * H_];
    __shared__ int   touched[GG_];

    const int i = blockIdx.x;
    const int t = threadIdx.x;   // 0..127 hidden dim

#pragma unroll
    for (int c = 0; c < GG_; ++c) maxbuf[c * H_ + t] = -__builtin_inff();
    if (t < GG_) touched[t] = 0;
    __syncthreads();

    const int   fi  = fpos[i];
    const float pix = pos[2 * i];
    const float piy = pos[2 * i + 1];

    if (fi) {
        for (int j = 0; j < NAG_; ++j) {
            if (j == i) continue;
            if (!vj[j]) continue;
            float rx = pos[2 * j]     - pix;
            float ry = pos[2 * j + 1] - piy;
            // in-box test (NaN rel fails automatically)
            if (!(fabsf(rx) <= NS_ * 0.5f) || !(fabsf(ry) <= NS_ * 0.5f)) continue;
            // cell = NS/G = 0.5; floor((rel + NS/2)/cell) == trunc((rel+1)*2) for rel in box
            int gx = (int)((rx + NS_ * 0.5f) * (G_ / NS_));
            int gy = (int)((ry + NS_ * 0.5f) * (G_ / NS_));
            gx = min(G_ - 1, max(0, gx));
            gy = min(G_ - 1, max(0, gy));
            int cell = gx * G_ + gy;

            float h = hidden[(size_t)j * H_ + t];
            float* p = &maxbuf[cell * H_ + t];   // thread t exclusively owns dim t
            *p = fmaxf(*p, h);
            if (t == 0) touched[cell] = 1;
        }
    }
    __syncthreads();

    float* orow = pooled + (size_t)i * FANIN_;
#pragma unroll
    for (int c = 0; c < GG_; ++c) {
        float v = touched[c] ? maxbuf[c * H_ + t] : 0.0f;
        orow[c * H_ + t] = v;
    }
}

// ---------------------------------------------------------------------------
// Kernel 2: out = relu(pooled @ W^T + b) via V_WMMA_F32_16X16X4_F32
//   One wave per 16x16 output tile. 64 M-tiles x 8 N-tiles = 512 waves.
//   A frag (16x4 f32, 2 VGPRs): lane<16 -> K={k,k+1}, lane>=16 -> K={k+2,k+3}
//   B frag (4x16 f32, 2 VGPRs): same K split; B[k][n] = W[colN][k]
//   C/D (8 VGPRs): vgpr r -> row = 16*tileM + r + 8*(lane>=16), col = 16*tileN + (lane&15)
// ---------------------------------------------------------------------------
__global__ void gemm_wmma_kernel(const float* __restrict__ A,   // pooled [1024,2048]
                                 const float* __restrict__ W,   // [128,2048]
                                 const float* __restrict__ bias,
                                 float* __restrict__ out) {     // [1024,128]
    const int wave  = (blockIdx.x * blockDim.x + threadIdx.x) >> 5;
    const int lane  = threadIdx.x & 31;
    const int tileM = wave >> 3;   // 0..63
    const int tileN = wave & 7;    // 0..7
    const int half  = lane >> 4;   // 0 or 1
    const int l16   = lane & 15;

    const float* arow = A + (size_t)(tileM * 16 + l16) * FANIN_ + half * 2;
    const float* wrow = W + (size_t)(tileN * 16 + l16) * FANIN_ + half * 2;

    v8f acc = {0.f, 0.f, 0.f, 0.f, 0.f, 0.f, 0.f, 0.f};

#pragma unroll 4
    for (int k = 0; k < FANIN_; k += 4) {
        v2f a = *(const v2f*)(arow + k);
        v2f b = *(const v2f*)(wrow + k);
        // 8 args: (neg_a, A, neg_b, B, c_mod, C, reuse_a, reuse_b)
        acc = __builtin_amdgcn_wmma_f32_16x16x4_f32(
            false, a, false, b, (short)0, acc, false, false);
    }

    const int   col = tileN * 16 + l16;
    const float bb  = bias[col];
#pragma unroll
    for (int r = 0; r < 8; ++r) {
        int   row = tileM * 16 + r + half * 8;
        float v   = acc[r] + bb;
        out[(size_t)row * POOL_ + col] = fmaxf(v, 0.0f);
    }
}

// ---------------------------------------------------------------------------
extern "C" void kernel_launch(void* const* d_in, const int* in_sizes, int n_in,
                              void* d_out, int out_size, void* d_ws, size_t ws_size,
                              hipStream_t stream) {
    (void)in_sizes; (void)n_in; (void)out_size; (void)ws_size;

    const float* pos    = (const float*)d_in[0];
    const float* hidden = (const float*)d_in[1];
    const float* W      = (const float*)d_in[2];
    const float* b      = (const float*)d_in[3];
    float*       out    = (float*)d_out;

    float* pooled = (float*)d_ws;                                   // 8 MB
    int*   fpos   = (int*)((char*)d_ws + (size_t)NAG_ * FANIN_ * sizeof(float));
    int*   vj     = fpos + NAG_;

    validity_kernel<<<(NAG_ + 255) / 256, 256, 0, stream>>>(pos, hidden, fpos, vj);
    pool_kernel<<<NAG_, H_, 0, stream>>>(pos, hidden, fpos, vj, pooled);
    gemm_wmma_kernel<<<128, 128, 0, stream>>>(pooled, W, b, out);
}